// MessagePassing_30812095381894
// MI455X (gfx1250) — compile-verified
//
#include <hip/hip_runtime.h>

// out = segment_sum(x[src], dst), N=100000 nodes, E=1.6M edges, D=64 f32 features.
//
// Memory-bound (102 MFLOP vs ~460MB traffic); working set is L2-resident on MI455X
// (192MB L2). Push/atomic costs 64 f32 L2 atomics per edge (102.4M RMWs) -> that,
// not bandwidth, is the wall. So: bucket edges by destination (count -> 3-phase
// parallel scan -> bucket; 3.2M int atomics total), then PULL: one half-wave
// (16 lanes x float4) per node reduces its bucket in registers and issues a single
// global_store_b128. Deterministic fallback to the atomic kernel if ws too small.

#define D_FEAT 64
#define SCAN_BLK 1024   // elements per scan block (power of 2)

// ---------------------------------------------------------------- utilities --

__global__ __launch_bounds__(256) void gnn_zero_i32_kernel(int* __restrict__ p, int n) {
    int i = blockIdx.x * 256 + threadIdx.x;
    if (i < n) p[i] = 0;
}

__global__ __launch_bounds__(256) void gnn_zero_f4_kernel(float4* __restrict__ out, int n4) {
    int i = blockIdx.x * 256 + threadIdx.x;
    if (i < n4) out[i] = make_float4(0.0f, 0.0f, 0.0f, 0.0f);
}

// wave32 inclusive scan in registers (no LDS, no barriers)
__device__ __forceinline__ int wave_incl_scan(int v, unsigned lane) {
    #pragma unroll
    for (int off = 1; off < 32; off <<= 1) {
        int t = __shfl_up(v, off, 32);
        if (lane >= (unsigned)off) v += t;
    }
    return v;
}

// ------------------------------------------------------- CSR (pull) pipeline --

// Pass 1: histogram of destinations (no-return global_atomic_add_u32).
__global__ __launch_bounds__(256) void gnn_count_kernel(
    const int* __restrict__ dst_idx, int* __restrict__ counts, int n_edges)
{
    int e = blockIdx.x * 256 + threadIdx.x;
    if (e >= n_edges) return;
    __builtin_prefetch(&dst_idx[e + 16384], 0, 0);
    int d = __builtin_nontemporal_load(dst_idx + e);
    atomicAdd(&counts[d], 1);
}

// Pass 2a: per-block exclusive scan (1024 elements/block, all blocks parallel).
// Shuffle scan within each wave, 32-entry LDS combine across the 32 waves.
__global__ __launch_bounds__(SCAN_BLK) void gnn_scan_local_kernel(
    const int* __restrict__ counts, int* __restrict__ offsets,
    int* __restrict__ bsums, int n)
{
    const unsigned tid  = threadIdx.x;
    const unsigned lane = tid & 31u;
    const unsigned wave = tid >> 5;           // 0..31
    const int i = blockIdx.x * SCAN_BLK + (int)tid;

    int v = (i < n) ? counts[i] : 0;
    int incl = wave_incl_scan(v, lane);       // inclusive within wave

    __shared__ int wsum[32];
    if (lane == 31u) wsum[wave] = incl;
    __syncthreads();
    if (wave == 0) {                          // scan the 32 wave totals
        int wv = wsum[lane];
        wsum[lane] = wave_incl_scan(wv, lane);
    }
    __syncthreads();

    int block_incl = incl + (wave > 0 ? wsum[wave - 1] : 0);
    if (i < n) offsets[i] = block_incl - v;   // block-local EXCLUSIVE scan
    if (tid == SCAN_BLK - 1) bsums[blockIdx.x] = block_incl;  // block total
}

// Pass 2b: exclusive scan of the (<=128) block totals, in place. One block.
__global__ __launch_bounds__(128) void gnn_scan_sums_kernel(int* __restrict__ bsums, int nb)
{
    const unsigned tid  = threadIdx.x;
    const unsigned lane = tid & 31u;
    const unsigned wave = tid >> 5;           // 0..3

    int v = ((int)tid < nb) ? bsums[tid] : 0;
    int incl = wave_incl_scan(v, lane);

    __shared__ int wsum[4];
    if (lane == 31u) wsum[wave] = incl;
    __syncthreads();
    if (tid == 0) {                           // exclusive scan of 4 wave totals
        int r = 0;
        #pragma unroll
        for (int k = 0; k < 4; ++k) { int t = wsum[k]; wsum[k] = r; r += t; }
    }
    __syncthreads();

    if ((int)tid < nb) bsums[tid] = incl - v + wsum[wave];   // global exclusive
}

// Pass 2c: add block offsets; materialize offsets and the cursor working copy.
__global__ __launch_bounds__(256) void gnn_scan_add_kernel(
    int* __restrict__ offsets, int* __restrict__ cursor,
    const int* __restrict__ bsums, int n)
{
    int i = blockIdx.x * 256 + threadIdx.x;
    if (i >= n) return;
    int o = offsets[i] + bsums[i >> 10];      // i / SCAN_BLK
    offsets[i] = o;
    cursor[i]  = o;
}

// Pass 3: bucket edges by destination; returning global_atomic_add_u32 gives slot.
__global__ __launch_bounds__(256) void gnn_bucket_kernel(
    const int* __restrict__ src_idx, const int* __restrict__ dst_idx,
    int* __restrict__ cursor, int* __restrict__ src_sorted, int n_edges)
{
    int e = blockIdx.x * 256 + threadIdx.x;
    if (e >= n_edges) return;
    __builtin_prefetch(&src_idx[e + 16384], 0, 0);
    __builtin_prefetch(&dst_idx[e + 16384], 0, 0);
    int s = __builtin_nontemporal_load(src_idx + e);
    int d = __builtin_nontemporal_load(dst_idx + e);
    int pos = atomicAdd(&cursor[d], 1);
    __builtin_nontemporal_store(s, src_sorted + pos);
}

// Pass 4: pull-reduce. Half-wave (16 lanes) per node, lane k owns features
// [4k,4k+4). Register accumulation; software-pipelined index load overlaps the
// current b128 row gather. One global_store_b128 per node; zero float atomics.
__global__ __launch_bounds__(256) void gnn_pull_kernel(
    const float* __restrict__ x, const int* __restrict__ src_sorted,
    const int* __restrict__ offsets, const int* __restrict__ counts,
    float* __restrict__ out, int n_nodes)
{
    const long long gid = (long long)blockIdx.x * 256 + threadIdx.x;
    const int node = (int)(gid >> 4);
    if (node >= n_nodes) return;
    const int c = ((int)gid & 15) << 2;       // 0,4,...,60

    const int beg = offsets[node];            // same-address across 16 lanes: coalesces
    const int end = beg + counts[node];

    float4 acc = make_float4(0.0f, 0.0f, 0.0f, 0.0f);
    int i = beg;
    if (i < end) {
        int s = src_sorted[i];
        for (; i + 1 < end; ++i) {
            int sn = src_sorted[i + 1];       // pipeline: overlaps the gather below
            const float4 v = *reinterpret_cast<const float4*>(
                x + ((long long)s * D_FEAT + c));
            acc.x += v.x; acc.y += v.y; acc.z += v.z; acc.w += v.w;
            s = sn;
        }
        const float4 v = *reinterpret_cast<const float4*>(
            x + ((long long)s * D_FEAT + c));
        acc.x += v.x; acc.y += v.y; acc.z += v.z; acc.w += v.w;
    }
    *reinterpret_cast<float4*>(out + ((long long)node * D_FEAT + c)) = acc;
}

// ------------------------------------------- fallback: direct atomic scatter --

__global__ __launch_bounds__(256) void gnn_scatter_atomic_kernel(
    const float* __restrict__ x,
    const int*   __restrict__ src_idx, const int* __restrict__ dst_idx,
    float*       __restrict__ out, int n_edges)
{
    const long long gid = (long long)blockIdx.x * 256 + threadIdx.x;
    const int e = (int)(gid >> 4);
    if (e >= n_edges) return;
    const int c = ((int)gid & 15) << 2;

    __builtin_prefetch(&src_idx[e + 16384], 0, 0);
    __builtin_prefetch(&dst_idx[e + 16384], 0, 0);
    const int s = __builtin_nontemporal_load(src_idx + e);
    const int d = __builtin_nontemporal_load(dst_idx + e);

    const float4 v = *reinterpret_cast<const float4*>(x + ((long long)s * D_FEAT + c));
    float* o = out + ((long long)d * D_FEAT + c);
    unsafeAtomicAdd(o + 0, v.x);
    unsafeAtomicAdd(o + 1, v.y);
    unsafeAtomicAdd(o + 2, v.z);
    unsafeAtomicAdd(o + 3, v.w);
}

// -------------------------------------------------------------------- launch --

extern "C" void kernel_launch(void* const* d_in, const int* in_sizes, int n_in,
                              void* d_out, int out_size, void* d_ws, size_t ws_size,
                              hipStream_t stream) {
    const float* x    = (const float*)d_in[0];
    const int*   edge = (const int*)d_in[1];      // flat (2,E): row 0 = src, row 1 = dst
    const int n_edges = in_sizes[1] / 2;
    const int* src = edge;
    const int* dst = edge + n_edges;
    float* out = (float*)d_out;
    const int n_nodes = out_size / D_FEAT;

    const int nb_scan = (n_nodes + SCAN_BLK - 1) / SCAN_BLK;   // 98 blocks; must be <=128
    const size_t needed =
        ((size_t)3 * n_nodes + (size_t)n_edges + (size_t)nb_scan) * sizeof(int);

    if (ws_size >= needed && nb_scan <= 128) {
        // ---- CSR pull pipeline: 3.2M int atomics instead of 102.4M f32 atomics
        int* counts     = (int*)d_ws;             // N
        int* offsets    = counts  + n_nodes;      // N
        int* cursor     = offsets + n_nodes;      // N
        int* src_sorted = cursor  + n_nodes;      // E
        int* bsums      = src_sorted + n_edges;   // nb_scan

        const int nb = (n_nodes + 255) / 256;
        const int eb = (n_edges + 255) / 256;

        gnn_zero_i32_kernel <<<nb, 256, 0, stream>>>(counts, n_nodes);
        gnn_count_kernel    <<<eb, 256, 0, stream>>>(dst, counts, n_edges);
        gnn_scan_local_kernel<<<nb_scan, SCAN_BLK, 0, stream>>>(counts, offsets, bsums, n_nodes);
        gnn_scan_sums_kernel <<<1, 128, 0, stream>>>(bsums, nb_scan);
        gnn_scan_add_kernel  <<<nb, 256, 0, stream>>>(offsets, cursor, bsums, n_nodes);
        gnn_bucket_kernel   <<<eb, 256, 0, stream>>>(src, dst, cursor, src_sorted, n_edges);

        const long long tot = (long long)n_nodes * 16;
        gnn_pull_kernel<<<(int)((tot + 255) / 256), 256, 0, stream>>>(
            x, src_sorted, offsets, counts, out, n_nodes);
    } else {
        // ---- fallback: zero output, then direct f32-atomic scatter
        const int n4 = out_size / 4;
        gnn_zero_f4_kernel<<<(n4 + 255) / 256, 256, 0, stream>>>((float4*)d_out, n4);

        const long long tot = (long long)n_edges * 16;
        gnn_scatter_atomic_kernel<<<(int)((tot + 255) / 256), 256, 0, stream>>>(
            x, src, dst, out, n_edges);
    }
}